// NNKernel_74071005986944
// MI455X (gfx1250) — compile-verified
//
#include <hip/hip_runtime.h>
#include <math.h>

typedef __attribute__((ext_vector_type(2))) float v2f;
typedef __attribute__((ext_vector_type(8))) float v8f;

#define N_ROWS   8192
#define KDIM     64
#define TILE     128          // block tile (M and N)
#define LDSS     68           // LDS row stride in floats (16B aligned, conflict-free)

__constant__ float kPI     = 3.14159265358979323846f;
__constant__ float kINV2PI = 0.15915494309189535f;
#define F32_TINY 1.17549435082228751e-38f

// ---------------------------------------------------------------------------
// Kernel 1: row statistics  xx[i] = 0.5*||x_i||^2 + 1  (and same for y)
// ---------------------------------------------------------------------------
__global__ __launch_bounds__(256) void row_stats_kernel(
    const float* __restrict__ x, const float* __restrict__ y,
    float* __restrict__ xx, float* __restrict__ yy)
{
    int i = blockIdx.x * blockDim.x + threadIdx.x;      // 0 .. 16383
    const float* src = (i < N_ROWS) ? (x + (size_t)i * KDIM)
                                    : (y + (size_t)(i - N_ROWS) * KDIM);
    const float4* p = (const float4*)src;
    float s = 0.0f;
#pragma unroll
    for (int j = 0; j < KDIM / 4; ++j) {
        float4 v = p[j];
        s += v.x * v.x + v.y * v.y + v.z * v.z + v.w * v.w;
    }
    float r = 0.5f * s + 1.0f;
    if (i < N_ROWS) xx[i] = r; else yy[i - N_ROWS] = r;
}

// ---------------------------------------------------------------------------
// Kernel 2: 128x128 LDS-tiled GEMM via V_WMMA_F32_16X16X4_F32 + arccos epilogue
// ---------------------------------------------------------------------------
__global__ __launch_bounds__(256) void arccos_wmma_kernel(
    const float* __restrict__ x, const float* __restrict__ y,
    const float* __restrict__ xx, const float* __restrict__ yy,
    float* __restrict__ out)
{
    __shared__ float Xs[TILE * LDSS];
    __shared__ float Ys[TILE * LDSS];

    const int t    = threadIdx.x;
    const int lane = t & 31;
    const int w    = t >> 5;            // wave id 0..7
    const int half = lane >> 4;         // 0: lanes 0-15, 1: lanes 16-31
    const int l16  = lane & 15;

    const int m0 = blockIdx.y * TILE;
    const int n0 = blockIdx.x * TILE;

    // ---- stage X/Y tiles (128 rows x 64 floats) into LDS, float4 granular ----
#pragma unroll
    for (int i = 0; i < 8; ++i) {
        int idx = i * 256 + t;          // 0..2047 float4 slots
        int row = idx >> 4;             // 16 float4 per row
        int c4  = idx & 15;
        float4 vx = *(const float4*)&x[(size_t)(m0 + row) * KDIM + c4 * 4];
        float4 vy = *(const float4*)&y[(size_t)(n0 + row) * KDIM + c4 * 4];
        *(float4*)&Xs[row * LDSS + c4 * 4] = vx;
        *(float4*)&Ys[row * LDSS + c4 * 4] = vy;
    }
    __syncthreads();

    // ---- each wave: 16 rows (r0..r0+15) x 128 cols (8 tiles of 16) ----
    const int r0 = w * 16;
    v8f acc[8];
#pragma unroll
    for (int nt = 0; nt < 8; ++nt) acc[nt] = v8f{0,0,0,0,0,0,0,0};

    const float* __restrict__ xrow = &Xs[(r0 + l16) * LDSS];

    for (int kb = 0; kb < KDIM; kb += 4) {
        const int ko = kb + half * 2;   // A 16x4 layout: lanes<16 hold K={0,1}, lanes>=16 K={2,3}
        v2f a;
        a.x = xrow[ko];
        a.y = xrow[ko + 1];
#pragma unroll
        for (int nt = 0; nt < 8; ++nt) {
            const float* __restrict__ ycol = &Ys[(nt * 16 + l16) * LDSS];
            v2f b;                      // B 4x16 mirrored layout
            b.x = ycol[ko];
            b.y = ycol[ko + 1];
            acc[nt] = __builtin_amdgcn_wmma_f32_16x16x4_f32(
                false, a, false, b, (short)0, acc[nt], false, false);
        }
    }

    // ---- epilogue: arc-cosine kernel math + non-temporal streaming stores ----
    const int gr_base = m0 + r0 + half * 8;  // C/D layout: VGPR r -> row r (+8 for hi half)
    float xxv[8];
#pragma unroll
    for (int r = 0; r < 8; ++r) xxv[r] = xx[gr_base + r];

#pragma unroll
    for (int nt = 0; nt < 8; ++nt) {
        const int gc = n0 + nt * 16 + l16;
        const float yv = yy[gc];
#pragma unroll
        for (int r = 0; r < 8; ++r) {
            float s     = acc[nt][r];
            float xyv   = 0.5f * s + 1.0f;
            float denom = xxv[r] * yv + F32_TINY;
            float ct    = fminf(fmaxf(xyv * rsqrtf(denom), -1.0f), 1.0f);
            float st    = sqrtf(fmaxf(denom - xyv * xyv, 0.0f));
            float th    = acosf(ct);
            float res   = (st + (kPI - th) * xyv) * kINV2PI;
            __builtin_nontemporal_store(res,
                &out[(size_t)(gr_base + r) * N_ROWS + gc]);
        }
    }
}

// ---------------------------------------------------------------------------
extern "C" void kernel_launch(void* const* d_in, const int* in_sizes, int n_in,
                              void* d_out, int out_size, void* d_ws, size_t ws_size,
                              hipStream_t stream)
{
    const float* x = (const float*)d_in[0];   // [8192, 64] f32
    const float* y = (const float*)d_in[1];   // [8192, 64] f32
    float* out = (float*)d_out;               // [8192, 8192] f32

    float* xx = (float*)d_ws;                 // 8192 f32
    float* yy = xx + N_ROWS;                  // 8192 f32  (64 KB total)

    // 1) row norms
    row_stats_kernel<<<(2 * N_ROWS) / 256, 256, 0, stream>>>(x, y, xx, yy);

    // 2) WMMA GEMM + epilogue: grid of 128x128 tiles
    dim3 grid(N_ROWS / TILE, N_ROWS / TILE);  // (64, 64)
    arccos_wmma_kernel<<<grid, 256, 0, stream>>>(x, y, xx, yy, out);
}